// GeneGNN_71459665871601
// MI455X (gfx1250) — compile-verified
//
#include <hip/hip_runtime.h>
#include <hip/hip_bf16.h>

#define N_NODES  50000
#define N_EDGES  800000
#define N_GRAPHS 64
#define HID      128
#define FEAT     64
#define OUTD     32
#define N_LAYERS 3

typedef __attribute__((ext_vector_type(2))) float v2f;
typedef __attribute__((ext_vector_type(8))) float v8f;

// ---------------- async staging: global A tile -> LDS (padded) ----------------
// Copies 16 rows x K floats of A (row-major, stride lda) into LDS laid out with
// row stride S floats (S = Ktot+4 pad: keeps B128 stores 16B-aligned and makes
// fragment reads bank-conflict-free). Uses CDNA5 async-to-LDS path (ASYNCcnt).
__device__ __forceinline__ void stage_async(const float* __restrict__ A, int lda,
                                            int rowBase, int K, int S, int koff,
                                            float* As) {
  const int kq4 = K >> 2;              // float4 chunks per row
  const int chunks = 16 * kq4;         // always a multiple of blockDim
  for (int c = threadIdx.x; c < chunks; c += blockDim.x) {
    int m = c / kq4, kq = c - m * kq4;
    unsigned laddr = (unsigned)(size_t)&As[m * S + koff + kq * 4];
    const float* g = A + (size_t)(rowBase + m) * (size_t)lda + kq * 4;
    asm volatile("global_load_async_to_lds_b128 %0, %1, off"
                 :: "v"(laddr), "v"(g)
                 : "memory");
  }
}

__device__ __forceinline__ void stage_wait() {
  asm volatile("s_wait_asynccnt 0" ::: "memory");
  __syncthreads();
}

// ---------------- WMMA core: one wave computes a 16x16 f32 tile ----------------
// A fragments from LDS (padded tile), B row-major K x N from global.
// f32 16x16x4 fragment layout (ISA 7.12.2):
//   A 16x4: v0 lanes0-15 K=0, lanes16-31 K=2 ; v1 lanes0-15 K=1, lanes16-31 K=3
//   B 4x16: v0 lanes0-15 row K=0 (N=lane), lanes16-31 row K=2 ; v1 rows 1,3
//   C/D 16x16: vgpr r -> lanes0-15 M=r, lanes16-31 M=8+r
__device__ __forceinline__ v8f wmma_lds_kloop(const float* As, int S, int koff, int K,
                                              const float* __restrict__ B, int ldb,
                                              int colBase, v8f acc) {
  const int lane = threadIdx.x & 31;
  const int hk   = (lane >> 4) << 1;   // 0 or 2
  const int m    = lane & 15;
  const float* arow = As + m * S + koff;
  const float* bcol = B + colBase + m;
  for (int k = 0; k < K; k += 4) {
    v2f a = *(const v2f*)&arow[k + hk];          // ds_load_b64, conflict-free
    v2f b;
    b.x = bcol[(size_t)(k + hk) * (size_t)ldb];
    b.y = bcol[(size_t)(k + hk + 1) * (size_t)ldb];
    acc = __builtin_amdgcn_wmma_f32_16x16x4_f32(false, a, false, b, (short)0, acc,
                                                false, false);
  }
  return acc;
}

// ---------------- GEMM kernels (256 thr = 8 waves -> 16 rows x 128 cols) -------

// h0 = x @ initW[0:64,:] + imp[row]*v1[col] + v0[col]
__global__ void k_gemm_h0(const float* __restrict__ x, const float* __restrict__ Wtop,
                          const float* __restrict__ v1, const float* __restrict__ v0,
                          const float* __restrict__ imp, float* __restrict__ h) {
  __shared__ float As[16 * (FEAT + 4)];
  int rowBase = blockIdx.x * 16;
  int colBase = (threadIdx.x >> 5) * 16;
  stage_async(x, FEAT, rowBase, FEAT, FEAT + 4, 0, As);
  stage_wait();
  v8f acc = {};
  acc = wmma_lds_kloop(As, FEAT + 4, 0, FEAT, Wtop, HID, colBase, acc);
  int lane = threadIdx.x & 31;
  int n = lane & 15, mo = (lane >> 4) * 8;
#pragma unroll
  for (int r = 0; r < 8; ++r) {
    int row = rowBase + mo + r, col = colBase + n;
    h[(size_t)row * HID + col] = acc[r] + imp[row] * v1[col] + v0[col];
  }
}

// C = [A1 | A2] @ B + bias   (B is (K1+K2) x 128 row-major; A2 may be null/K2==0)
__global__ void k_gemm_cb(const float* __restrict__ A1, int K1,
                          const float* __restrict__ A2, int K2,
                          const float* __restrict__ B, const float* __restrict__ bias,
                          float* __restrict__ C) {
  __shared__ float As[16 * (2 * HID + 4)];
  int rowBase = blockIdx.x * 16;
  int colBase = (threadIdx.x >> 5) * 16;
  int S = K1 + K2 + 4;
  stage_async(A1, K1, rowBase, K1, S, 0, As);
  if (K2 > 0) stage_async(A2, K2, rowBase, K2, S, K1, As);
  stage_wait();
  v8f acc = {};
  acc = wmma_lds_kloop(As, S, 0, K1, B, HID, colBase, acc);
  if (K2 > 0)
    acc = wmma_lds_kloop(As, S, K1, K2, B + (size_t)K1 * HID, HID, colBase, acc);
  int lane = threadIdx.x & 31;
  int n = lane & 15, mo = (lane >> 4) * 8;
#pragma unroll
  for (int r = 0; r < 8; ++r) {
    int row = rowBase + mo + r, col = colBase + n;
    C[(size_t)row * HID + col] = acc[r] + bias[col];
  }
}

// gate = sigmoid(conv @ gW[0:128,:] + imp[row]*gW[128,:] + gb); out = g*conv+(1-g)*h
__global__ void k_gemm_gate(const float* __restrict__ conv, const float* __restrict__ gW,
                            const float* __restrict__ gb, const float* __restrict__ imp,
                            const float* __restrict__ h, float* __restrict__ outb) {
  __shared__ float As[16 * (HID + 4)];
  int rowBase = blockIdx.x * 16;
  int colBase = (threadIdx.x >> 5) * 16;
  const int S = HID + 4;
  stage_async(conv, HID, rowBase, HID, S, 0, As);
  stage_wait();
  v8f acc = {};
  acc = wmma_lds_kloop(As, S, 0, HID, gW, HID, colBase, acc);
  int lane = threadIdx.x & 31;
  int n = lane & 15, mo = (lane >> 4) * 8;
  const float* gWlast = gW + (size_t)HID * HID;   // row 128 of (129x128)
#pragma unroll
  for (int r = 0; r < 8; ++r) {
    int row = rowBase + mo + r, col = colBase + n;
    float z = acc[r] + imp[row] * gWlast[col] + gb[col];
    float g = 1.0f / (1.0f + __expf(-z));
    float cv = As[(mo + r) * S + col];            // conv tile already in LDS
    outb[(size_t)row * HID + col] = g * cv + (1.0f - g) * h[(size_t)row * HID + col];
  }
}

// hf = h @ finW + finB ; atomic segment-sum into logits[batch[row]]  (64 thr = 2 waves)
__global__ void k_gemm_final(const float* __restrict__ h, const float* __restrict__ fW,
                             const float* __restrict__ fb, const int* __restrict__ batch,
                             float* __restrict__ logits) {
  __shared__ float As[16 * (HID + 4)];
  int rowBase = blockIdx.x * 16;
  int colBase = (threadIdx.x >> 5) * 16;
  const int S = HID + 4;
  stage_async(h, HID, rowBase, HID, S, 0, As);
  stage_wait();
  v8f acc = {};
  acc = wmma_lds_kloop(As, S, 0, HID, fW, OUTD, colBase, acc);
  int lane = threadIdx.x & 31;
  int n = lane & 15, mo = (lane >> 4) * 8;
#pragma unroll
  for (int r = 0; r < 8; ++r) {
    int row = rowBase + mo + r, col = colBase + n;
    atomicAdd(&logits[(size_t)batch[row] * OUTD + col], acc[r] + fb[col]);
  }
}

// ---------------- helpers / elementwise --------------------------------------

__global__ void k_zero(float* p, size_t n) {
  for (size_t i = blockIdx.x * (size_t)blockDim.x + threadIdx.x; i < n;
       i += (size_t)gridDim.x * blockDim.x)
    p[i] = 0.0f;
}

__device__ __forceinline__ unsigned f2o(float f) {
  unsigned u = __float_as_uint(f);
  return (u & 0x80000000u) ? ~u : (u | 0x80000000u);
}
__device__ __forceinline__ float o2f(unsigned u) {
  return (u & 0x80000000u) ? __uint_as_float(u & 0x7FFFFFFFu) : __uint_as_float(~u);
}

__global__ void k_init_mm(unsigned* mn, unsigned* mx) {
  int i = threadIdx.x;
  if (i < N_GRAPHS) { mn[i] = 0xFFFFFFFFu; mx[i] = 0u; }
}

__global__ void k_minmax(const float* __restrict__ v, const float* __restrict__ add,
                         const int* __restrict__ batch, unsigned* mn, unsigned* mx) {
  int n = blockIdx.x * blockDim.x + threadIdx.x;
  if (n >= N_NODES) return;
  float val = v[n] + (add ? add[n] : 0.0f);
  unsigned o = f2o(val);
  int g = batch[n];
  atomicMin(&mn[g], o);
  atomicMax(&mx[g], o);
}

__global__ void k_norm(const float* __restrict__ v, const float* __restrict__ add,
                       const unsigned* __restrict__ mn, const unsigned* __restrict__ mx,
                       const int* __restrict__ batch, float* __restrict__ dst, int thresh) {
  int n = blockIdx.x * blockDim.x + threadIdx.x;
  if (n >= N_NODES) return;
  float val = v[n] + (add ? add[n] : 0.0f);
  int g = batch[n];
  float lo = o2f(mn[g]), hi = o2f(mx[g]);
  float o = (val - lo) / (hi - lo + 1e-8f);
  if (thresh && o < 0.1f) o = 0.0f;
  dst[n] = o;
}

__global__ void k_counts(const int* __restrict__ batch, float* counts) {
  int n = blockIdx.x * blockDim.x + threadIdx.x;
  if (n < N_NODES) atomicAdd(&counts[batch[n]], 1.0f);
}

// v1[o] = sum_j impW[j]*initW[64+j][o]; v0[o] = sum_j impB[j]*initW[64+j][o] + initB[o]
__global__ void k_v01(const float* __restrict__ impW, const float* __restrict__ impB,
                      const float* __restrict__ initW, const float* __restrict__ initB,
                      float* v1, float* v0) {
  int o = threadIdx.x;
  if (o >= HID) return;
  float s1 = 0.0f, s0 = 0.0f;
  for (int j = 0; j < HID; ++j) {
    float w = initW[(size_t)(FEAT + j) * HID + o];
    s1 += impW[j] * w;
    s0 += impB[j] * w;
  }
  v1[o] = s1;
  v0[o] = s0 + initB[o];
}

// one wave per edge: aggr[dst] += hW[src]  (128 floats, float4 per lane)
__global__ void k_scatter(const float* __restrict__ hW, const int* __restrict__ src,
                          const int* __restrict__ dst, float* __restrict__ aggr) {
  int e = (blockIdx.x * blockDim.x + threadIdx.x) >> 5;
  int lane = threadIdx.x & 31;
  if (e >= N_EDGES) return;
  int s = src[e], d = dst[e];
  float4 v = ((const float4*)(hW + (size_t)s * HID))[lane];
  float* p = aggr + (size_t)d * HID + lane * 4;
  atomicAdd(p + 0, v.x); atomicAdd(p + 1, v.y);
  atomicAdd(p + 2, v.z); atomicAdd(p + 3, v.w);
}

// one wave per node: prop = out.pW + pb ; tmp1 = 0.9*imp + 0.1*prop
__global__ void k_prop(const float* __restrict__ outb, const float* __restrict__ pW,
                       const float* __restrict__ pb, const float* __restrict__ imp,
                       float* __restrict__ tmp1) {
  int node = blockIdx.x * 8 + (threadIdx.x >> 5);
  int lane = threadIdx.x & 31;
  if (node >= N_NODES) return;
  const float* row = outb + (size_t)node * HID;
  float s = 0.0f;
#pragma unroll
  for (int j = 0; j < 4; ++j) s += row[lane + 32 * j] * pW[lane + 32 * j];
#pragma unroll
  for (int off = 16; off > 0; off >>= 1) s += __shfl_xor(s, off);
  if (lane == 0) {
    float p = s + pb[0];
    tmp1[node] = 0.9f * imp[node] + (1.0f - 0.9f) * p;
  }
}

// graph-norm: sums of out per (graph,col)
__global__ void k_gn_sum(const float* __restrict__ outb, const int* __restrict__ batch,
                         float* __restrict__ sums) {
  int t = blockIdx.x * blockDim.x + threadIdx.x;  // N_NODES*32 threads
  int node = t >> 5;
  if (node >= N_NODES) return;
  int cg = (t & 31) * 4;
  int g = batch[node];
  float4 v = *(const float4*)(outb + (size_t)node * HID + cg);
  float* p = sums + (size_t)g * HID + cg;
  atomicAdd(p + 0, v.x); atomicAdd(p + 1, v.y);
  atomicAdd(p + 2, v.z); atomicAdd(p + 3, v.w);
}

__global__ void k_gn_finmean(float* mean, const float* counts) {
  int i = blockIdx.x * blockDim.x + threadIdx.x;
  if (i < N_GRAPHS * HID) mean[i] /= counts[i >> 7];
}

__global__ void k_gn_var(const float* __restrict__ outb, const int* __restrict__ batch,
                         const float* __restrict__ mean, const float* __restrict__ a,
                         float* __restrict__ var) {
  size_t i = blockIdx.x * (size_t)blockDim.x + threadIdx.x;  // N_NODES*HID
  if (i >= (size_t)N_NODES * HID) return;
  int node = (int)(i >> 7), c = (int)(i & 127);
  int g = batch[node];
  float t = outb[i] - a[c] * mean[(size_t)g * HID + c];
  atomicAdd(&var[(size_t)g * HID + c], t * t);
}

__global__ void k_gn_finistd(float* var, const float* counts) {
  int i = blockIdx.x * blockDim.x + threadIdx.x;
  if (i < N_GRAPHS * HID) var[i] = rsqrtf(var[i] / counts[i >> 7] + 1e-5f);
}

__global__ void k_gn_apply(const float* __restrict__ outb, const int* __restrict__ batch,
                           const float* __restrict__ mean, const float* __restrict__ istd,
                           const float* __restrict__ w, const float* __restrict__ b,
                           const float* __restrict__ a, float* __restrict__ hn) {
  size_t i = blockIdx.x * (size_t)blockDim.x + threadIdx.x;
  if (i >= (size_t)N_NODES * HID) return;
  int node = (int)(i >> 7), c = (int)(i & 127);
  int g = batch[node];
  float cen = outb[i] - a[c] * mean[(size_t)g * HID + c];
  float r = w[c] * cen * istd[(size_t)g * HID + c] + b[c];
  hn[i] = r > 0.0f ? r : 0.0f;
}

__global__ void k_logits_fin(const float* __restrict__ logits, const float* __restrict__ counts,
                             float* __restrict__ out) {
  int i = blockIdx.x * blockDim.x + threadIdx.x;
  if (i < N_GRAPHS * OUTD) out[i] = logits[i] / counts[i >> 5];
}

// ---------------- host side ---------------------------------------------------

extern "C" void kernel_launch(void* const* d_in, const int* in_sizes, int n_in,
                              void* d_out, int out_size, void* d_ws, size_t ws_size,
                              hipStream_t stream) {
  (void)in_sizes; (void)n_in; (void)out_size; (void)ws_size;
  const float* x     = (const float*)d_in[0];
  const float* imp0  = (const float*)d_in[2];   // initial_importance (N,1)
  const float* impW  = (const float*)d_in[3];
  const float* impB  = (const float*)d_in[4];
  const float* initW = (const float*)d_in[5];
  const float* initB = (const float*)d_in[6];
  const float* msgW  = (const float*)d_in[7];
  const float* msgB  = (const float*)d_in[8];
  const float* updW  = (const float*)d_in[9];
  const float* updB  = (const float*)d_in[10];
  const float* gateW = (const float*)d_in[11];
  const float* gateB = (const float*)d_in[12];
  const float* propW = (const float*)d_in[13];
  const float* propB = (const float*)d_in[14];
  const float* normw = (const float*)d_in[15];
  const float* normb = (const float*)d_in[16];
  const float* norma = (const float*)d_in[17];
  const float* finW  = (const float*)d_in[18];
  const float* finB  = (const float*)d_in[19];
  const int*   eidx  = (const int*)d_in[20];
  const int*   batch = (const int*)d_in[21];
  const int* srcI = eidx;
  const int* dstI = eidx + N_EDGES;

  float* ws = (float*)d_ws;
  const size_t NH = (size_t)N_NODES * HID;
  float* bufA  = ws;
  float* bufB  = ws + NH;
  float* aggr  = ws + 2 * NH;
  float* convb = ws + 3 * NH;
  float* outb  = ws + 4 * NH;
  float* impv  = ws + 5 * NH;
  float* tmp1  = impv + N_NODES;
  float* tmp2  = tmp1 + N_NODES;
  float* v1    = tmp2 + N_NODES;
  float* v0    = v1 + HID;
  float* counts = v0 + HID;
  unsigned* mnO = (unsigned*)(counts + N_GRAPHS);
  unsigned* mxO = mnO + N_GRAPHS;
  float* mean   = (float*)(mxO + N_GRAPHS);
  float* var    = mean + N_GRAPHS * HID;
  float* logits = var + N_GRAPHS * HID;

  float* outp = (float*)d_out;

  const int GEMM_BLOCKS = N_NODES / 16;          // 3125
  const int NODE_BLOCKS = (N_NODES + 255) / 256; // 196
  const int GN_EL_BLOCKS = (int)(NH / 256);      // 25000
  const int GN_SUM_BLOCKS = N_NODES * 32 / 256;  // 6250

  // counts (once)
  k_zero<<<1, 256, 0, stream>>>(counts, N_GRAPHS);
  k_counts<<<NODE_BLOCKS, 256, 0, stream>>>(batch, counts);

  // importance = normalize(imp0)
  k_init_mm<<<1, 64, 0, stream>>>(mnO, mxO);
  k_minmax<<<NODE_BLOCKS, 256, 0, stream>>>(imp0, nullptr, batch, mnO, mxO);
  k_norm<<<NODE_BLOCKS, 256, 0, stream>>>(imp0, nullptr, mnO, mxO, batch, impv, 0);

  // h0
  k_v01<<<1, HID, 0, stream>>>(impW, impB, initW, initB, v1, v0);
  k_gemm_h0<<<GEMM_BLOCKS, 256, 0, stream>>>(x, initW, v1, v0, impv, bufA);

  float* hcur = bufA;
  float* hnext = bufB;

  for (int l = 0; l < N_LAYERS; ++l) {
    const float* mW = msgW + (size_t)l * HID * HID;
    const float* mB = msgB + (size_t)l * HID;
    const float* uW = updW + (size_t)l * 2 * HID * HID;
    const float* uB = updB + (size_t)l * HID;
    const float* gW = gateW + (size_t)l * (HID + 1) * HID;
    const float* gB = gateB + (size_t)l * HID;
    const float* pW = propW + (size_t)l * HID;
    const float* pB = propB + l;
    const float* nw = normw + (size_t)l * HID;
    const float* nb = normb + (size_t)l * HID;
    const float* na = norma + (size_t)l * HID;
    float* hWbuf = hnext;   // free until graph-norm apply reuses it as h_next

    // hW = h @ msg_W + msg_b  (bias-per-edge folds into node-side bias)
    k_gemm_cb<<<GEMM_BLOCKS, 256, 0, stream>>>(hcur, HID, nullptr, 0, mW, mB, hWbuf);

    // aggr = scatter-sum over edges
    k_zero<<<4096, 256, 0, stream>>>(aggr, NH);
    k_scatter<<<N_EDGES / 8, 256, 0, stream>>>(hWbuf, srcI, dstI, aggr);

    // conv = [h | aggr] @ upd_W + upd_b
    k_gemm_cb<<<GEMM_BLOCKS, 256, 0, stream>>>(hcur, HID, aggr, HID, uW, uB, convb);

    // gate + blend -> outb
    k_gemm_gate<<<GEMM_BLOCKS, 256, 0, stream>>>(convb, gW, gB, impv, hcur, outb);

    // prop -> tmp1 = 0.9*imp + 0.1*prop
    k_prop<<<N_NODES / 8, 256, 0, stream>>>(outb, pW, pB, impv, tmp1);

    // importance: normalize, threshold, normalize
    k_init_mm<<<1, 64, 0, stream>>>(mnO, mxO);
    k_minmax<<<NODE_BLOCKS, 256, 0, stream>>>(tmp1, nullptr, batch, mnO, mxO);
    k_norm<<<NODE_BLOCKS, 256, 0, stream>>>(tmp1, nullptr, mnO, mxO, batch, tmp2, 1);
    k_init_mm<<<1, 64, 0, stream>>>(mnO, mxO);
    k_minmax<<<NODE_BLOCKS, 256, 0, stream>>>(tmp2, nullptr, batch, mnO, mxO);
    k_norm<<<NODE_BLOCKS, 256, 0, stream>>>(tmp2, nullptr, mnO, mxO, batch, impv, 0);

    // graph norm + relu -> h_next (reuse hW buffer)
    k_zero<<<32, 256, 0, stream>>>(mean, N_GRAPHS * HID);
    k_gn_sum<<<GN_SUM_BLOCKS, 256, 0, stream>>>(outb, batch, mean);
    k_gn_finmean<<<32, 256, 0, stream>>>(mean, counts);
    k_zero<<<32, 256, 0, stream>>>(var, N_GRAPHS * HID);
    k_gn_var<<<GN_EL_BLOCKS, 256, 0, stream>>>(outb, batch, mean, na, var);
    k_gn_finistd<<<32, 256, 0, stream>>>(var, counts);
    k_gn_apply<<<GN_EL_BLOCKS, 256, 0, stream>>>(outb, batch, mean, var, nw, nb, na, hWbuf);

    float* t = hcur; hcur = hnext; hnext = t;
  }

  // logits = segment_mean(h @ final_W + final_b)
  k_zero<<<8, 256, 0, stream>>>(logits, N_GRAPHS * OUTD);
  k_gemm_final<<<GEMM_BLOCKS, 64, 0, stream>>>(hcur, finW, finB, batch, logits);
  k_logits_fin<<<8, 256, 0, stream>>>(logits, counts, outp);

  // final_importance = normalize(imp0 + importance)
  k_init_mm<<<1, 64, 0, stream>>>(mnO, mxO);
  k_minmax<<<NODE_BLOCKS, 256, 0, stream>>>(impv, imp0, batch, mnO, mxO);
  k_norm<<<NODE_BLOCKS, 256, 0, stream>>>(impv, imp0, mnO, mxO, batch,
                                          outp + N_GRAPHS * OUTD, 0);
}